// TimeSliceEncoder_16578573762772
// MI455X (gfx1250) — compile-verified
//
#include <hip/hip_runtime.h>
#include <stdint.h>

#ifndef __has_builtin
#define __has_builtin(x) 0
#endif

#define BLOCK 256
#define EPT 4                       // events per thread per stage
#define STAGE (BLOCK * EPT)         // 1024 events = 16KB per stage buffer
#define TW 50.0f
#define SLICE_DUR 10.0f
#define NSLICES 5
#define DOWN_W 80
#define DOWN_H 60
#define OUT_ELEMS (NSLICES * 2 * DOWN_H * DOWN_W)   // 48000

typedef int v4i __attribute__((ext_vector_type(4)));
typedef __attribute__((address_space(1))) v4i g_v4i;   // global
typedef __attribute__((address_space(3))) v4i l_v4i;   // LDS
typedef __attribute__((address_space(3))) void lds_void_t;

// ---- monotone float<->uint key (unsigned order == float order) ----
__device__ __forceinline__ unsigned fkey(float f) {
  unsigned b = __float_as_uint(f);
  return (b & 0x80000000u) ? ~b : (b | 0x80000000u);
}
__device__ __forceinline__ float funkey(unsigned k) {
  unsigned b = (k & 0x80000000u) ? (k ^ 0x80000000u) : ~k;
  return __uint_as_float(b);
}

// ---- CDNA5 async global->LDS copy (16B per lane), ASYNCcnt tracked ----
__device__ __forceinline__ void cp_async16(void* lds, const void* g) {
#if __has_builtin(__builtin_amdgcn_global_load_async_to_lds_b128)
  __builtin_amdgcn_global_load_async_to_lds_b128(
      (g_v4i*)(void*)g, (l_v4i*)lds, 0, 0);
#else
  unsigned l = (unsigned)(uintptr_t)(lds_void_t*)lds;
  asm volatile("global_load_async_to_lds_b128 %0, %1, off"
               :: "v"(l), "v"(g) : "memory");
#endif
}

template <int N>
__device__ __forceinline__ void wait_asynccnt() {
#if __has_builtin(__builtin_amdgcn_s_wait_asynccnt)
  __builtin_amdgcn_s_wait_asynccnt(N);
#else
  asm volatile("s_wait_asynccnt %0" :: "i"(N) : "memory");
#endif
}

template <int N>
__device__ __forceinline__ void wait_dscnt() {
#if __has_builtin(__builtin_amdgcn_s_wait_dscnt)
  __builtin_amdgcn_s_wait_dscnt(N);
#else
  asm volatile("s_wait_dscnt %0" :: "i"(N) : "memory");
#endif
}

// ----------------- kernel 1: init output + workspace -----------------
__global__ void ts_init(float* __restrict__ out, int out_n, unsigned* __restrict__ ws) {
  int i = blockIdx.x * blockDim.x + threadIdx.x;
  if (i == 0) { ws[0] = 0xFFFFFFFFu; ws[1] = 0u; }  // min-key, max-key
  for (int j = i; j < out_n; j += gridDim.x * blockDim.x) out[j] = 0.0f;
}

// ----------------- kernel 2: min/max of t column -----------------
__global__ void ts_minmax(const float4* __restrict__ ev, int nev, unsigned* __restrict__ ws) {
  unsigned kmin = 0xFFFFFFFFu, kmax = 0u;
  long long stride = (long long)gridDim.x * blockDim.x;
  for (long long i = (long long)blockIdx.x * blockDim.x + threadIdx.x; i < nev; i += stride) {
    float4 e = ev[i];
    unsigned k = fkey(e.z);
    kmin = min(kmin, k);
    kmax = max(kmax, k);
  }
  // wave32 reduction
  for (int off = 16; off > 0; off >>= 1) {
    kmin = min(kmin, (unsigned)__shfl_xor((int)kmin, off, 32));
    kmax = max(kmax, (unsigned)__shfl_xor((int)kmax, off, 32));
  }
  __shared__ unsigned smin[8], smax[8];
  int lane = threadIdx.x & 31, w = threadIdx.x >> 5;
  if (lane == 0) { smin[w] = kmin; smax[w] = kmax; }
  __syncthreads();
  if (threadIdx.x == 0) {
    unsigned m = smin[0], M = smax[0];
    int nw = blockDim.x >> 5;
    for (int j = 1; j < nw; ++j) { m = min(m, smin[j]); M = max(M, smax[j]); }
    atomicMin(ws + 0, m);
    atomicMax(ws + 1, M);
  }
}

// ----------------- per-event binning (matches reference IEEE ops) -----------------
__device__ __forceinline__ void process_event(float4 e, float tmin, float dclamp,
                                              bool donorm, float* __restrict__ out) {
  float tt = e.z;
  if (donorm) {
    tt = tt - tmin;
    tt = tt / dclamp;     // IEEE div, same as jnp
    tt = tt * TW;
  }
  int sl = (int)(tt / SLICE_DUR);                 // trunc toward zero, like astype(int32)
  sl = sl < 0 ? 0 : (sl > NSLICES - 1 ? NSLICES - 1 : sl);
  int xi = (int)(e.x * 0.125f);                   // /8 exact as *0.125
  xi = xi < 0 ? 0 : (xi > DOWN_W - 1 ? DOWN_W - 1 : xi);
  int yi = (int)(e.y * 0.125f);
  yi = yi < 0 ? 0 : (yi > DOWN_H - 1 ? DOWN_H - 1 : yi);
  int ch = sl * 2 + (e.w <= 0.0f ? 1 : 0);
  out[ch * (DOWN_H * DOWN_W) + yi * DOWN_W + xi] = 1.0f;
}

// ----------------- kernel 3: async-LDS staged scatter -----------------
__global__ void __launch_bounds__(BLOCK)
ts_scatter(const float4* __restrict__ ev, int nev,
           const unsigned* __restrict__ ws, float* __restrict__ out) {
  const float tmin = funkey(ws[0]);
  const float tmax = funkey(ws[1]);
  const float denom = tmax - tmin;
  const float dclamp = fmaxf(denom, 1e-12f);
  const bool donorm = denom > 0.0f;

  __shared__ float4 buf[2][STAGE];   // 32 KB double buffer
  const int t = threadIdx.x;
  const int numStages = nev / STAGE;
  const int tailStart = numStages * STAGE;

  // walk stages in DESCENDING address order: pass-1 leaves the tail of the
  // 256MB stream resident in the 192MB L2, so reverse traversal re-hits it.
  auto issue = [&](int sIdx, int b) {
    wait_dscnt<0>();                               // prior LDS reads retired (WAR)
    const float4* src = ev + (size_t)(numStages - 1 - sIdx) * STAGE;
#pragma unroll
    for (int k = 0; k < EPT; ++k) {
      int idx = k * BLOCK + t;                     // lanes contiguous per issue
      cp_async16(&buf[b][idx], src + idx);
    }
  };

  int s = blockIdx.x;
  int cur = 0;
  if (s < numStages) issue(s, 0);
  for (; s < numStages; s += gridDim.x) {
    int nxt = s + gridDim.x;
    if (nxt < numStages) {
      issue(nxt, cur ^ 1);
      wait_asynccnt<EPT>();                        // oldest stage landed (in-order)
    } else {
      wait_asynccnt<0>();
    }
    asm volatile("" ::: "memory");
#pragma unroll
    for (int k = 0; k < EPT; ++k) {
      float4 e = buf[cur][k * BLOCK + t];          // each thread reads its own slots
      process_event(e, tmin, dclamp, donorm, out);
    }
    cur ^= 1;
  }

  // tail (none for 16.7M events, but stay general)
  for (long long i = tailStart + (long long)blockIdx.x * BLOCK + t; i < nev;
       i += (long long)gridDim.x * BLOCK) {
    process_event(ev[i], tmin, dclamp, donorm, out);
  }
}

extern "C" void kernel_launch(void* const* d_in, const int* in_sizes, int n_in,
                              void* d_out, int out_size, void* d_ws, size_t ws_size,
                              hipStream_t stream) {
  const float* ev = (const float*)d_in[0];
  const int nev = in_sizes[0] / 4;
  float* out = (float*)d_out;
  unsigned* ws = (unsigned*)d_ws;

  ts_init<<<dim3((out_size + 255) / 256), dim3(256), 0, stream>>>(out, out_size, ws);
  ts_minmax<<<dim3(4096), dim3(256), 0, stream>>>((const float4*)ev, nev, ws);
  ts_scatter<<<dim3(4096), dim3(BLOCK), 0, stream>>>((const float4*)ev, nev, ws, out);
}